// GraphConvolution2_22660247453735
// MI455X (gfx1250) — compile-verified
//
#include <hip/hip_runtime.h>

#define D_FEAT      128
#define TILE_EDGES  256
#define BLOCK       256
#define WAVES_PER_BLOCK (BLOCK / 32)

// ---- CDNA5 async global->LDS path (probe via __has_builtin) ----------------
#if defined(__HIP_DEVICE_COMPILE__) && \
    __has_builtin(__builtin_amdgcn_global_load_async_to_lds_b32)
#define USE_ASYNC_LDS 1
#else
#define USE_ASYNC_LDS 0
#endif

// Builtin expects typed, non-const address-space pointers:
//   param0: int addrspace(1)*  (global source)
//   param1: int addrspace(3)*  (LDS destination)
typedef __attribute__((address_space(1))) int* gint_p;  // global int*
typedef __attribute__((address_space(3))) int* lint_p;  // LDS int*

__device__ __forceinline__ void wait_async_zero() {
#if defined(__HIP_DEVICE_COMPILE__)
#if __has_builtin(__builtin_amdgcn_s_wait_asynccnt)
  __builtin_amdgcn_s_wait_asynccnt(0);
#else
  asm volatile("s_wait_asynccnt 0" ::: "memory");
#endif
#endif
}

// ---------------------------------------------------------------------------
// Kernel 1: out = alpha * feature   (fully initializes d_out; float4 = b128)
// ---------------------------------------------------------------------------
__global__ __launch_bounds__(BLOCK) void gcn_combine_init(
    const float* __restrict__ feature,
    const float* __restrict__ alpha_p,
    float* __restrict__ out,
    int n4) {
  const int i = blockIdx.x * BLOCK + threadIdx.x;
  if (i >= n4) return;
  const float a = *alpha_p;                       // uniform scalar load
  float4 v = ((const float4*)feature)[i];         // global_load_b128
  float4 r;
  r.x = a * v.x; r.y = a * v.y; r.z = a * v.z; r.w = a * v.w;
  ((float4*)out)[i] = r;                          // global_store_b128
}

// ---------------------------------------------------------------------------
// Kernel 2: scatter-add  out[row] += (1-alpha)*val * x[col]
//   - 256-edge tile staged into LDS via async copy engine (ASYNCcnt)
//   - one wave per edge; lane l handles floats [4l, 4l+4)  (wave32: 4*32=128)
//   - 4x global_atomic_add_f32 per lane, non-returning, resolved in L2
// ---------------------------------------------------------------------------
__global__ __launch_bounds__(BLOCK) void gcn_spmm_scatter(
    const float* __restrict__ x,
    const int*   __restrict__ adj_row,
    const int*   __restrict__ adj_col,
    const float* __restrict__ adj_val,
    const float* __restrict__ alpha_p,
    float* __restrict__ out,
    int n_edges) {
  __shared__ int   s_row[TILE_EDGES];
  __shared__ int   s_col[TILE_EDGES];
  __shared__ float s_val[TILE_EDGES];

  const int tid       = threadIdx.x;
  const int lane      = tid & 31;
  const int wave      = tid >> 5;
  const int tile_base = blockIdx.x * TILE_EDGES;
  const int e         = tile_base + tid;

#if USE_ASYNC_LDS
  if (e < n_edges) {
    __builtin_amdgcn_global_load_async_to_lds_b32(
        (gint_p)(adj_row + e), (lint_p)&s_row[tid], 0, 0);
    __builtin_amdgcn_global_load_async_to_lds_b32(
        (gint_p)(adj_col + e), (lint_p)&s_col[tid], 0, 0);
    __builtin_amdgcn_global_load_async_to_lds_b32(
        (gint_p)(adj_val + e), (lint_p)&s_val[tid], 0, 0);
  }
  wait_async_zero();          // this wave's async copies complete
#else
  if (e < n_edges) {
    s_row[tid] = adj_row[e];
    s_col[tid] = adj_col[e];
    s_val[tid] = adj_val[e];
  }
#endif
  __syncthreads();            // all waves' staged tuples visible

  const float scale = 1.0f - *alpha_p;
  const int rem        = n_edges - tile_base;
  const int tile_count = (rem < TILE_EDGES) ? rem : TILE_EDGES;

  for (int i = wave; i < tile_count; i += WAVES_PER_BLOCK) {
    const int   row = s_row[i];           // ds_load_b32 (broadcast)
    const int   col = s_col[i];
    const float w   = s_val[i] * scale;

    const float4 xv = *(const float4*)(x + (size_t)col * D_FEAT + lane * 4);
    float* dst = out + (size_t)row * D_FEAT + lane * 4;

    unsafeAtomicAdd(dst + 0, w * xv.x);   // global_atomic_add_f32 (no return)
    unsafeAtomicAdd(dst + 1, w * xv.y);
    unsafeAtomicAdd(dst + 2, w * xv.z);
    unsafeAtomicAdd(dst + 3, w * xv.w);
  }
}

// ---------------------------------------------------------------------------
extern "C" void kernel_launch(void* const* d_in, const int* in_sizes, int n_in,
                              void* d_out, int out_size, void* d_ws, size_t ws_size,
                              hipStream_t stream) {
  const float* feature = (const float*)d_in[0];
  const float* x       = (const float*)d_in[1];
  const int*   adj_row = (const int*)d_in[2];
  const int*   adj_col = (const int*)d_in[3];
  const float* adj_val = (const float*)d_in[4];
  const float* alpha_p = (const float*)d_in[5];   // device scalar

  const int n_feat_total = in_sizes[0];           // N_NODES * D_FEAT
  const int n_edges      = in_sizes[2];
  const int n4           = n_feat_total / 4;

  float* out = (float*)d_out;

  gcn_combine_init<<<dim3((n4 + BLOCK - 1) / BLOCK), dim3(BLOCK), 0, stream>>>(
      feature, alpha_p, out, n4);

  gcn_spmm_scatter<<<dim3((n_edges + TILE_EDGES - 1) / TILE_EDGES),
                     dim3(BLOCK), 0, stream>>>(
      x, adj_row, adj_col, adj_val, alpha_p, out, n_edges);
}